// PointNet2_6682969112986
// MI455X (gfx1250) — compile-verified
//
#include <hip/hip_runtime.h>

typedef __attribute__((ext_vector_type(2))) float v2f;
typedef __attribute__((ext_vector_type(8))) float v8f;
typedef unsigned int u32x4 __attribute__((ext_vector_type(4)));
typedef int i32x8 __attribute__((ext_vector_type(8)));
typedef int i32x4 __attribute__((ext_vector_type(4)));

#define TPB 256
#define KCHUNK 512  // K-floats per LDS-resident W panel chunk (32KB)

static __device__ __forceinline__ v2f ld2(const float* p) {
  return *(const v2f*)p;  // 8B-aligned by construction
}

// ---------------------------------------------------------------------------
// GEMM via V_WMMA_F32_16X16X4_F32 with TDM-staged weights.
// X:[R,K4] row-major (K4 = align4(K), pad cols zero).
// Wp packed [K4/4][C][4] -> the 16-col panel for this block is a 2D tile of
// 64 contiguous floats per k-group with row stride C*4 floats: wave 0 pulls it
// into LDS with tensor_load_to_lds (TENSORcnt), then all 8 waves read B
// fragments from LDS (ds_load_b64) while streaming A from global.
// Block = 8 waves x 32 rows = 256 rows x 16 cols.  R % 256 == 0 for all layers.
// Epilogue writes per-32-row-group (sum,sumsq) partials for deterministic BN.
// ---------------------------------------------------------------------------
__global__ void gemm_wmma_kernel(const float* __restrict__ X, const float* __restrict__ Wp,
                                 const float* __restrict__ bias, float* __restrict__ H,
                                 float* __restrict__ partial, int R, int K4, int C) {
  extern __shared__ float panel[];  // (KCHUNK/4)*64 floats; LDS offset 0 (no static LDS)
  const int lane = threadIdx.x & 31;
  const int wave = threadIdx.x >> 5;  // 8 waves
  const int colTiles = C >> 4;
  const int rowBlock = (int)(blockIdx.x / colTiles);
  const int tn = ((int)(blockIdx.x % colTiles)) << 4;
  const int tm = (rowBlock << 8) + (wave << 5);
  const int half = lane >> 4;
  const int l16 = lane & 15;
  const float* xrow0 = X + (size_t)(tm + l16) * K4 + 2 * half;
  const float* xrow1 = xrow0 + (size_t)16 * K4;
  __builtin_prefetch(xrow0, 0, 0);  // global_prefetch_b8
  __builtin_prefetch(xrow1, 0, 0);
  v8f acc0 = {}, acc1 = {};
  const float* bp = panel + l16 * 4 + 2 * half;
  for (int kc = 0; kc < K4; kc += KCHUNK) {
    const int kChunk = (K4 - kc < KCHUNK) ? (K4 - kc) : KCHUNK;
    const int kgN = kChunk >> 2;
    __syncthreads();  // previous panel fully consumed
#if __has_builtin(__builtin_amdgcn_tensor_load_to_lds)
    if (wave == 0) {
      // ---- D# group 0: count=1 | lds_addr | global_addr | type=2 ----
      const unsigned long long ga =
          (unsigned long long)(const char*)(Wp + ((size_t)(kc >> 2) * C + tn) * 4);
      u32x4 g0;
      g0.x = 1u;                                    // count=1, user descriptor
      g0.y = 0u;                                    // lds_addr = dynamic-LDS base
      g0.z = (unsigned)(ga & 0xffffffffu);          // global_addr[31:0]
      g0.w = (unsigned)((ga >> 32) & 0x01ffffffu) | (2u << 30);  // ga[56:32] | type=2
      // ---- D# group 1: mask=0, data_size=4B, dims/strides of the 2D tile ----
      const unsigned d0 = 64u;            // tensor_dim0 (floats per packed row)
      const unsigned d1 = (unsigned)kgN;  // tensor_dim1 = #k-groups
      const unsigned s0 = (unsigned)C * 4u;  // tensor_dim0_stride (floats)
      i32x8 g1;
      g1[0] = (int)(2u << 16);                               // data_size=4B
      g1[1] = (int)((d0 & 0xffffu) << 16);                   // tensor_dim0[15:0]
      g1[2] = (int)(((d0 >> 16) & 0xffffu) | ((d1 & 0xffffu) << 16));
      g1[3] = (int)(((d1 >> 16) & 0xffffu) | ((d0 & 0xffffu) << 16));  // | tile_dim0
      g1[4] = (int)(d1 & 0xffffu);                           // tile_dim1 (tile_dim2=0)
      g1[5] = (int)s0;                                       // tensor_dim0_stride[31:0]
      g1[6] = 0;                                             // stride hi | dim1_stride lo
      g1[7] = 0;
      const i32x4 z4 = {0, 0, 0, 0};
      const i32x8 z8 = {0, 0, 0, 0, 0, 0, 0, 0};
      // 6-arg form (clang-23 / therock-10.0 headers): groups 2/3 + extra group
      // operand are NULL/zero for a 2D tensor tile.
      __builtin_amdgcn_tensor_load_to_lds(g0, g1, z4, z4, z8, 0);
      __builtin_amdgcn_s_wait_tensorcnt(0);
    }
#else
    for (int i = threadIdx.x; i < (kChunk << 4); i += blockDim.x) {
      const int kg = i >> 6, w = i & 63;
      panel[i] = Wp[((size_t)((kc >> 2) + kg) * C + tn) * 4 + w];
    }
#endif
    __syncthreads();  // panel visible to all waves
    for (int kg = 0; kg < kgN; ++kg) {
      const v2f a0 = ld2(xrow0 + kc + kg * 4);
      const v2f a1 = ld2(xrow1 + kc + kg * 4);
      const v2f b  = *(const v2f*)(bp + kg * 64);  // ds_load_b64
      acc0 = __builtin_amdgcn_wmma_f32_16x16x4_f32(false, a0, false, b, (short)0, acc0,
                                                   false, false);
      acc1 = __builtin_amdgcn_wmma_f32_16x16x4_f32(false, a1, false, b, (short)0, acc1,
                                                   false, false);
    }
  }
  const float bc = bias[tn + l16];
  float s1 = 0.0f, s2 = 0.0f;
#pragma unroll
  for (int j = 0; j < 8; ++j) {
    const float v0 = acc0[j] + bc;
    const float v1 = acc1[j] + bc;
    H[(size_t)(tm + j + 8 * half) * C + (tn + l16)]      = v0;
    H[(size_t)(tm + 16 + j + 8 * half) * C + (tn + l16)] = v1;
    s1 += v0 + v1;
    s2 += v0 * v0 + v1 * v1;
  }
  s1 += __shfl_xor(s1, 16, 32);
  s2 += __shfl_xor(s2, 16, 32);
  if (half == 0) {
    const size_t p = ((size_t)(tm >> 5) * C + (tn + l16)) * 2;
    partial[p]     = s1;
    partial[p + 1] = s2;
  }
}

// Pack W[K,C] -> Wp[(K4/4)][C][4] with zero pad for k >= K.
__global__ void pack_w_kernel(const float* __restrict__ W, float* __restrict__ Wp,
                              int K, int C, long long total /*K4*C*/) {
  long long t = (long long)blockIdx.x * blockDim.x + threadIdx.x;
  if (t >= total) return;
  const int j = (int)(t & 3);
  const long long q = t >> 2;
  const int c = (int)(q % C);
  const int kg = (int)(q / C);
  const int k = kg * 4 + j;
  Wp[t] = (k < K) ? W[(size_t)k * C + c] : 0.0f;
}

// Deterministic reduction of per-row-group partials -> per-channel (sum, sumsq)
__global__ void stats_reduce_kernel(const float* __restrict__ partial, float* __restrict__ stats,
                                    int rowGroups, int C) {
  int c = blockIdx.x * blockDim.x + threadIdx.x;
  if (c >= C) return;
  float s1 = 0.0f, s2 = 0.0f;
  for (int r = 0; r < rowGroups; ++r) {
    s1 += partial[((size_t)r * C + c) * 2];
    s2 += partial[((size_t)r * C + c) * 2 + 1];
  }
  stats[2 * c]     = s1;
  stats[2 * c + 1] = s2;
}

// y = relu((h - mu) * rsqrt(var + 1e-5) * g + beta), in place
__global__ void bn_relu_kernel(float* __restrict__ H, const float* __restrict__ stats,
                               const float* __restrict__ g, const float* __restrict__ beta,
                               long long total, int C, float invR) {
  long long t = (long long)blockIdx.x * blockDim.x + threadIdx.x;
  if (t >= total) return;
  const int c = (int)(t % C);
  const float mu  = stats[2 * c] * invR;
  const float var = stats[2 * c + 1] * invR - mu * mu;
  float x = (H[t] - mu) * rsqrtf(var + 1e-5f);
  x = x * g[c] + beta[c];
  H[t] = fmaxf(x, 0.0f);
}

// ---------------------------------------------------------------------------
// Farthest point sampling: one block per batch, register-resident distances.
// ---------------------------------------------------------------------------
__global__ void fps_kernel(const float* __restrict__ xyz, int* __restrict__ out,
                           int N, int npoint) {
  const int b = blockIdx.x;
  const int tid = threadIdx.x;
  const int T = blockDim.x;  // 256
  const float* P = xyz + (size_t)b * N * 3;
  float dloc[32];
  {
    int k = 0;
    for (int i = tid; i < N; i += T, ++k) dloc[k] = 1e10f;
  }
  __shared__ float ctr[3];
  __shared__ float rv[TPB];
  __shared__ int   ri[TPB];
  __shared__ int   farS;
  if (tid == 0) farS = 0;
  __syncthreads();
  for (int it = 0; it < npoint; ++it) {
    const int far = farS;
    if (tid == 0) {
      out[b * npoint + it] = far;
      ctr[0] = P[3 * far];
      ctr[1] = P[3 * far + 1];
      ctr[2] = P[3 * far + 2];
    }
    __syncthreads();
    const float cx = ctr[0], cy = ctr[1], cz = ctr[2];
    float bv = -1.0f;
    int   bi = 0;
    int k = 0;
    for (int i = tid; i < N; i += T, ++k) {
      const float dx = P[3 * i] - cx, dy = P[3 * i + 1] - cy, dz = P[3 * i + 2] - cz;
      const float d = dx * dx + dy * dy + dz * dz;
      const float m = fminf(dloc[k], d);
      dloc[k] = m;
      if (m > bv) { bv = m; bi = i; }
    }
    rv[tid] = bv;
    ri[tid] = bi;
    __syncthreads();
    for (int s = T >> 1; s > 0; s >>= 1) {
      if (tid < s) {
        if (rv[tid + s] > rv[tid] || (rv[tid + s] == rv[tid] && ri[tid + s] < ri[tid])) {
          rv[tid] = rv[tid + s];
          ri[tid] = ri[tid + s];
        }
      }
      __syncthreads();
    }
    if (tid == 0) farS = ri[0];
    __syncthreads();
  }
}

// Scan points in index order, keep first `ns` within radius, pad with first.
__global__ void ball_query_kernel(const float* __restrict__ xyz, const float* __restrict__ nxyz,
                                  int* __restrict__ idx, int N, int S, int ns, float r2,
                                  long long total) {
  long long t = (long long)blockIdx.x * blockDim.x + threadIdx.x;
  if (t >= total) return;
  const int b = (int)(t / S);
  const float* P = xyz + (size_t)b * N * 3;
  const float cx = nxyz[t * 3], cy = nxyz[t * 3 + 1], cz = nxyz[t * 3 + 2];
  int* o = idx + (size_t)t * ns;
  int cnt = 0, first = 0;
  for (int i = 0; i < N && cnt < ns; ++i) {
    const float dx = P[3 * i] - cx, dy = P[3 * i + 1] - cy, dz = P[3 * i + 2] - cz;
    if (dx * dx + dy * dy + dz * dz <= r2) {
      if (cnt == 0) first = i;
      o[cnt++] = i;
    }
  }
  for (int j = cnt; j < ns; ++j) o[j] = first;
}

// out row stride = `stride` (>= 3+Cp, pad zero-filled)
__global__ void group_kernel(const float* __restrict__ xyz, const float* __restrict__ pts,
                             const float* __restrict__ nxyz, const int* __restrict__ idx,
                             float* __restrict__ out, int N, int S, int ns, int Cp,
                             int stride, long long total) {
  long long t = (long long)blockIdx.x * blockDim.x + threadIdx.x;
  if (t >= total) return;
  const long long bs = t / ns;
  const int b = (int)(bs / S);
  const int id = idx[t];
  const float* p  = xyz + ((size_t)b * N + id) * 3;
  const float* nz = nxyz + (size_t)bs * 3;
  float* o = out + (size_t)t * stride;
  o[0] = p[0] - nz[0];
  o[1] = p[1] - nz[1];
  o[2] = p[2] - nz[2];
  const float* f = pts + ((size_t)b * N + id) * Cp;
  for (int c = 0; c < Cp; ++c) o[3 + c] = f[c];
  for (int c = 3 + Cp; c < stride; ++c) o[c] = 0.0f;
}

__global__ void gather3_kernel(const float* __restrict__ xyz, const int* __restrict__ idx,
                               float* __restrict__ out, int N, int S, long long total) {
  long long t = (long long)blockIdx.x * blockDim.x + threadIdx.x;
  if (t >= total) return;
  const int b = (int)(t / S);
  const int id = idx[t];
  const float* p = xyz + ((size_t)b * N + id) * 3;
  out[t * 3] = p[0];
  out[t * 3 + 1] = p[1];
  out[t * 3 + 2] = p[2];
}

__global__ void maxpool_kernel(const float* __restrict__ H, float* __restrict__ out,
                               long long total, int ns, int C) {
  long long t = (long long)blockIdx.x * blockDim.x + threadIdx.x;
  if (t >= total) return;
  const int c = (int)(t % C);
  const long long bs = t / C;
  const float* h = H + (size_t)bs * ns * C + c;
  float m = h[0];
  for (int k = 1; k < ns; ++k) m = fmaxf(m, h[(size_t)k * C]);
  out[t] = m;
}

__global__ void extract3_kernel(const float* __restrict__ in6, float* __restrict__ out3,
                                long long total) {
  long long t = (long long)blockIdx.x * blockDim.x + threadIdx.x;
  if (t >= total) return;
  out3[t * 3]     = in6[t * 6];
  out3[t * 3 + 1] = in6[t * 6 + 1];
  out3[t * 3 + 2] = in6[t * 6 + 2];
}

// sa3 feat row (stride 260): [l2_xyz(3) | l2_pts(256) | pad(1)]
__global__ void sa3_build_kernel(const float* __restrict__ l2xyz, const float* __restrict__ l2p,
                                 float* __restrict__ out, long long total) {
  long long t = (long long)blockIdx.x * blockDim.x + threadIdx.x;
  if (t >= total) return;
  const long long r = t / 260;
  const int c = (int)(t % 260);
  float v;
  if (c < 3)        v = l2xyz[r * 3 + c];
  else if (c < 259) v = l2p[r * 256 + (c - 3)];
  else              v = 0.0f;
  out[t] = v;
}

// fp3 feat row (stride 1280): [l2_pts(256) | broadcast l3_pts(1024)]
__global__ void fp3_build_kernel(const float* __restrict__ l2p, const float* __restrict__ l3p,
                                 float* __restrict__ out, long long total) {
  long long t = (long long)blockIdx.x * blockDim.x + threadIdx.x;
  if (t >= total) return;
  const long long r = t / 1280;
  const int c = (int)(t % 1280);
  out[t] = (c < 256) ? l2p[r * 256 + c] : l3p[(r / 128) * 1024 + (c - 256)];
}

__global__ void copy_cols_kernel(const float* __restrict__ src, float* __restrict__ dst,
                                 long long total, int C, int stride, int off) {
  long long t = (long long)blockIdx.x * blockDim.x + threadIdx.x;
  if (t >= total) return;
  const long long r = t / C;
  const int c = (int)(t % C);
  dst[r * stride + off + c] = src[t];
}

__global__ void copyf_kernel(const float* __restrict__ src, float* __restrict__ dst,
                             long long n) {
  long long t = (long long)blockIdx.x * blockDim.x + threadIdx.x;
  if (t < n) dst[t] = src[t];
}

// 3-NN inverse-distance interpolation (matches top_k(-d,3) + 1/(d+1e-8) weights)
__global__ void three_interp_kernel(const float* __restrict__ xyz1, const float* __restrict__ xyz2,
                                    const float* __restrict__ pts2, float* __restrict__ out,
                                    int N1, int S2, int C2, int stride, int off,
                                    long long total) {
  long long t = (long long)blockIdx.x * blockDim.x + threadIdx.x;
  if (t >= total) return;
  const int b = (int)(t / N1);
  const float qx = xyz1[t * 3], qy = xyz1[t * 3 + 1], qz = xyz1[t * 3 + 2];
  const float* P = xyz2 + (size_t)b * S2 * 3;
  float d0 = 1e30f, d1 = 1e30f, d2 = 1e30f;
  int i0 = 0, i1 = 0, i2 = 0;
  for (int j = 0; j < S2; ++j) {
    const float dx = qx - P[3 * j], dy = qy - P[3 * j + 1], dz = qz - P[3 * j + 2];
    const float d = dx * dx + dy * dy + dz * dz;
    if (d < d0)      { d2 = d1; i2 = i1; d1 = d0; i1 = i0; d0 = d; i0 = j; }
    else if (d < d1) { d2 = d1; i2 = i1; d1 = d; i1 = j; }
    else if (d < d2) { d2 = d; i2 = j; }
  }
  float w0 = 1.0f / (d0 + 1e-8f), w1 = 1.0f / (d1 + 1e-8f), w2 = 1.0f / (d2 + 1e-8f);
  const float ws = w0 + w1 + w2;
  w0 /= ws; w1 /= ws; w2 /= ws;
  const float* p0 = pts2 + ((size_t)b * S2 + i0) * C2;
  const float* p1 = pts2 + ((size_t)b * S2 + i1) * C2;
  const float* p2 = pts2 + ((size_t)b * S2 + i2) * C2;
  float* o = out + (size_t)t * stride + off;
  for (int c = 0; c < C2; ++c) o[c] = w0 * p0[c] + w1 * p1[c] + w2 * p2[c];
}

// logits[b,o,n] = h[b,n,:]·W[:,o] + bias[o]  (Cout=5, scalar path)
__global__ void head2_kernel(const float* __restrict__ h, const float* __restrict__ W,
                             const float* __restrict__ bias, float* __restrict__ out,
                             int N, int Cin, int Cout, long long total) {
  long long t = (long long)blockIdx.x * blockDim.x + threadIdx.x;
  if (t >= total) return;
  const int o = (int)(t % Cout);
  const long long bn = t / Cout;
  const int n = (int)(bn % N);
  const int b = (int)(bn / N);
  const float* x = h + (size_t)bn * Cin;
  float acc = bias[o];
  for (int k = 0; k < Cin; ++k) acc += x[k] * W[k * Cout + o];
  out[((size_t)b * Cout + o) * N + n] = acc;
}

// ---------------------------------------------------------------------------
// Host side
// ---------------------------------------------------------------------------
namespace {
struct Layer { const float *W, *b, *beta, *g; int Cin, Cout; };

inline unsigned gblk(long long n) { return (unsigned)((n + TPB - 1) / TPB); }
inline int align4(int k) { return (k + 3) & ~3; }
}  // namespace

extern "C" void kernel_launch(void* const* d_in, const int* in_sizes, int n_in,
                              void* d_out, int out_size, void* d_ws, size_t ws_size,
                              hipStream_t stream) {
  constexpr int B = 8, N = 8192;
  // ---- locate xyz by its unique flat size, params are the rest (JAX order) ----
  int xyz_i = 0;
  for (int i = 0; i < n_in; ++i)
    if (in_sizes[i] == B * N * 6) { xyz_i = i; break; }
  const float* XYZ6 = (const float*)d_in[xyz_i];
  const float* leaves[80];
  int nleaves = 0;
  for (int i = 0; i < n_in; ++i)
    if (i != xyz_i && nleaves < 80) leaves[nleaves++] = (const float*)d_in[i];

  int p = 0;
  auto take = [&](int cin, int cout) -> Layer {
    Layer l;
    l.W = leaves[p]; l.b = leaves[p + 1]; l.beta = leaves[p + 2]; l.g = leaves[p + 3];
    p += 4; l.Cin = cin; l.Cout = cout;
    return l;
  };
  // sorted-key order: fp1, fp2, fp3, head1, head2, sa1, sa2, sa3
  Layer fp1L[3]; fp1L[0] = take(128, 128); fp1L[1] = take(128, 128); fp1L[2] = take(128, 128);
  Layer fp2L[2]; fp2L[0] = take(384, 256); fp2L[1] = take(256, 128);
  Layer fp3L[2]; fp3L[0] = take(1280, 256); fp3L[1] = take(256, 256);
  Layer head1 = take(128, 128);
  const float* W2 = leaves[p]; const float* b2 = leaves[p + 1]; p += 2;
  Layer sa1L[3]; sa1L[0] = take(9, 64);   sa1L[1] = take(64, 64);   sa1L[2] = take(64, 128);
  Layer sa2L[3]; sa2L[0] = take(131, 128); sa2L[1] = take(128, 128); sa2L[2] = take(128, 256);
  Layer sa3L[3]; sa3L[0] = take(259, 256); sa3L[1] = take(256, 512); sa3L[2] = take(512, 1024);

  // ---- workspace layout, 256B aligned ----
  size_t off = 0;
  auto alloc = [&](size_t bytes) -> char* {
    char* q = (char*)d_ws + off;
    off += (bytes + 255) & ~(size_t)255;
    return q;
  };
  float* l0_xyz   = (float*)alloc((size_t)B * N * 3 * 4);
  int*   fpsIdx1  = (int*)alloc((size_t)B * 512 * 4);
  float* l1_xyz   = (float*)alloc((size_t)B * 512 * 3 * 4);
  int*   ballIdx1 = (int*)alloc((size_t)B * 512 * 64 * 4);
  float* l1_pts   = (float*)alloc((size_t)B * 512 * 128 * 4);
  int*   fpsIdx2  = (int*)alloc((size_t)B * 128 * 4);
  float* l2_xyz   = (float*)alloc((size_t)B * 128 * 3 * 4);
  int*   ballIdx2 = (int*)alloc((size_t)B * 128 * 64 * 4);
  float* l2_pts   = (float*)alloc((size_t)B * 128 * 256 * 4);
  float* l3_pts   = (float*)alloc((size_t)B * 1024 * 4);
  float* l2pN     = (float*)alloc((size_t)B * 128 * 256 * 4);
  float* l1pN     = (float*)alloc((size_t)B * 512 * 128 * 4);
  float* stats    = (float*)alloc((size_t)2 * 1024 * 4);
  float* Wpacked  = (float*)alloc((size_t)512 * 1024 * 4);       // max K4*C (sa3 L3)
  float* partial  = (float*)alloc((size_t)8192 * 1024 * 2 * 4);  // max rowGroups*C*2
  float* G        = (float*)alloc((size_t)65536 * 132 * 4);      // grouped/concat arena
  float* bufA     = (float*)alloc((size_t)262144 * 128 * 4);
  float* bufB     = (float*)alloc((size_t)262144 * 128 * 4);
  (void)ws_size; (void)out_size;

  // conv_bn_relu chain runner (ping-pong bufA/bufB, deterministic BN stats).
  // Input X must have row stride align4(Ls[0].Cin), zero-padded.
  auto run_mlp = [&](const float* X, long long R, const Layer* Ls, int nl) -> const float* {
    const float* in = X;
    float* outs[2] = {bufA, bufB};
    int w = 0;
    for (int i = 0; i < nl; ++i) {
      const Layer& l = Ls[i];
      float* H = outs[w];
      const int K4 = align4(l.Cin);
      pack_w_kernel<<<gblk((long long)K4 * l.Cout), TPB, 0, stream>>>(
          l.W, Wpacked, l.Cin, l.Cout, (long long)K4 * l.Cout);
      const unsigned blocks = (unsigned)((R >> 8) * (l.Cout >> 4));
      const size_t shmem = (size_t)(KCHUNK / 4) * 64 * 4;  // 32KB W panel
      gemm_wmma_kernel<<<blocks, TPB, shmem, stream>>>(in, Wpacked, l.b, H, partial,
                                                       (int)R, K4, l.Cout);
      stats_reduce_kernel<<<gblk(l.Cout), TPB, 0, stream>>>(partial, stats, (int)(R >> 5),
                                                            l.Cout);
      bn_relu_kernel<<<gblk(R * l.Cout), TPB, 0, stream>>>(H, stats, l.g, l.beta,
                                                           R * l.Cout, l.Cout,
                                                           1.0f / (float)R);
      in = H;
      w ^= 1;
    }
    return in;
  };

  // ---- l0 ----
  extract3_kernel<<<gblk((long long)B * N), TPB, 0, stream>>>(XYZ6, l0_xyz, (long long)B * N);

  // ---- sa1: 8192 -> 512 centers, ball r=0.2 ns=64, mlp 9->64->64->128, maxpool ----
  fps_kernel<<<B, TPB, 0, stream>>>(l0_xyz, fpsIdx1, N, 512);
  gather3_kernel<<<gblk((long long)B * 512), TPB, 0, stream>>>(l0_xyz, fpsIdx1, l1_xyz, N, 512,
                                                               (long long)B * 512);
  ball_query_kernel<<<gblk((long long)B * 512), TPB, 0, stream>>>(l0_xyz, l1_xyz, ballIdx1, N,
                                                                  512, 64, 0.04f,
                                                                  (long long)B * 512);
  group_kernel<<<gblk((long long)B * 512 * 64), TPB, 0, stream>>>(
      l0_xyz, XYZ6, l1_xyz, ballIdx1, G, N, 512, 64, 6, 12, (long long)B * 512 * 64);
  const float* h = run_mlp(G, (long long)B * 512 * 64, sa1L, 3);
  maxpool_kernel<<<gblk((long long)B * 512 * 128), TPB, 0, stream>>>(h, l1_pts,
                                                                     (long long)B * 512 * 128,
                                                                     64, 128);

  // ---- sa2: 512 -> 128 centers, ball r=0.4 ns=64, mlp 131->128->128->256 ----
  fps_kernel<<<B, TPB, 0, stream>>>(l1_xyz, fpsIdx2, 512, 128);
  gather3_kernel<<<gblk((long long)B * 128), TPB, 0, stream>>>(l1_xyz, fpsIdx2, l2_xyz, 512,
                                                               128, (long long)B * 128);
  ball_query_kernel<<<gblk((long long)B * 128), TPB, 0, stream>>>(l1_xyz, l2_xyz, ballIdx2,
                                                                  512, 128, 64, 0.16f,
                                                                  (long long)B * 128);
  group_kernel<<<gblk((long long)B * 128 * 64), TPB, 0, stream>>>(
      l1_xyz, l1_pts, l2_xyz, ballIdx2, G, 512, 128, 64, 128, 132,
      (long long)B * 128 * 64);
  h = run_mlp(G, (long long)B * 128 * 64, sa2L, 3);
  maxpool_kernel<<<gblk((long long)B * 128 * 256), TPB, 0, stream>>>(h, l2_pts,
                                                                     (long long)B * 128 * 256,
                                                                     64, 256);

  // ---- sa3: global group (B,1,128,259->260), mlp 259->256->512->1024, maxpool ----
  sa3_build_kernel<<<gblk((long long)B * 128 * 260), TPB, 0, stream>>>(
      l2_xyz, l2_pts, G, (long long)B * 128 * 260);
  h = run_mlp(G, (long long)B * 128, sa3L, 3);
  maxpool_kernel<<<gblk((long long)B * 1024), TPB, 0, stream>>>(h, l3_pts, (long long)B * 1024,
                                                                128, 1024);

  // ---- fp3: broadcast l3 over 128 pts, concat with l2_pts -> 1280, mlp -> 256 ----
  fp3_build_kernel<<<gblk((long long)B * 128 * 1280), TPB, 0, stream>>>(
      l2_pts, l3_pts, G, (long long)B * 128 * 1280);
  h = run_mlp(G, (long long)B * 128, fp3L, 2);
  copyf_kernel<<<gblk((long long)B * 128 * 256), TPB, 0, stream>>>(h, l2pN,
                                                                   (long long)B * 128 * 256);

  // ---- fp2: 3-NN interp (512 <- 128), concat with l1_pts -> 384, mlp -> 128 ----
  copy_cols_kernel<<<gblk((long long)B * 512 * 128), TPB, 0, stream>>>(
      l1_pts, G, (long long)B * 512 * 128, 128, 384, 0);
  three_interp_kernel<<<gblk((long long)B * 512), TPB, 0, stream>>>(
      l1_xyz, l2_xyz, l2pN, G, 512, 128, 256, 384, 128, (long long)B * 512);
  h = run_mlp(G, (long long)B * 512, fp2L, 2);
  copyf_kernel<<<gblk((long long)B * 512 * 128), TPB, 0, stream>>>(h, l1pN,
                                                                   (long long)B * 512 * 128);

  // ---- fp1: 3-NN interp (8192 <- 512) -> 128, mlp 128->128->128->128, then head1 ----
  three_interp_kernel<<<gblk((long long)B * N), TPB, 0, stream>>>(
      l0_xyz, l1_xyz, l1pN, G, N, 512, 128, 128, 0, (long long)B * N);
  Layer chain[4] = {fp1L[0], fp1L[1], fp1L[2], head1};
  h = run_mlp(G, (long long)B * N, chain, 4);

  // ---- head2 + outputs: logits (B,5,N) then l3_pts (B,1,1024) ----
  float* out = (float*)d_out;
  head2_kernel<<<gblk((long long)B * N * 5), TPB, 0, stream>>>(h, W2, b2, out, N, 128, 5,
                                                               (long long)B * N * 5);
  copyf_kernel<<<gblk((long long)B * 1024), TPB, 0, stream>>>(l3_pts,
                                                              out + (size_t)B * 5 * N,
                                                              (long long)B * 1024);
}